// RAPiD_68281390072482
// MI455X (gfx1250) — compile-verified
//
#include <hip/hip_runtime.h>

// RAPiD decode for MI455X (gfx1250): pure streaming elementwise kernel.
// Bandwidth-bound (151 MB @ 23.3 TB/s ~ 6.5us floor); no matmul structure,
// so the optimal CDNA5 path is b128 vector traffic with NT temporal hints,
// wave32 blocks, and inline v_exp_f32 / v_rcp_f32 transcendentals.

typedef float v4f __attribute__((ext_vector_type(4)));

#define LOG2E 1.4426950408889634f

__device__ __forceinline__ float fast_exp(float v) {
    return __builtin_amdgcn_exp2f(v * LOG2E);            // v_exp_f32
}
__device__ __forceinline__ float fast_sigmoid(float v) {
    // 1 / (1 + e^-v)
    return __builtin_amdgcn_rcpf(1.0f + __builtin_amdgcn_exp2f(-v * LOG2E));
}

// raw:     (nB=64, 18, 128, 128) f32,  channel c = a*6 + e
// anchors: (3, 2) f32
// out:     (64, 3*128*128, 6) f32
__global__ __launch_bounds__(256)
void rapid_decode_kernel(const float* __restrict__ raw,
                         const float* __restrict__ anchors,
                         const int*   __restrict__ img_h_p,
                         const int*   __restrict__ img_w_p,
                         float* __restrict__ out)
{
    constexpr int nW    = 128;
    constexpr int nH    = 128;
    constexpr int nA    = 3;
    constexpr int plane = nH * nW;               // 16384 floats / field plane

    // quad id: each thread handles 4 consecutive x positions
    const int q = blockIdx.x * blockDim.x + threadIdx.x;

    const int xq = q & (nW / 4 - 1);             // 0..31
    int rem      = q >> 5;
    const int y  = rem & (nH - 1);               // 0..127
    rem        >>= 7;
    const int a  = rem % nA;
    const int b  = rem / nA;
    const int x0 = xq << 2;

    // uniform scalar loads (KMcnt path)
    const float sx = (float)img_w_p[0] * (1.0f / (float)nW);   // 8.0
    const float sy = (float)img_h_p[0] * (1.0f / (float)nH);   // 8.0
    const float aw = anchors[2 * a + 0];
    const float ah = anchors[2 * a + 1];

    // ---- load 6 field planes, 4 x-positions each: 6x global_load_b128 NT ----
    const float* src = raw + ((size_t)(b * 18 + a * 6) * plane + (size_t)y * nW + x0);
    v4f v[6];
#pragma unroll
    for (int e = 0; e < 6; ++e) {
        v[e] = __builtin_nontemporal_load((const v4f*)(src + (size_t)e * plane));
    }

    // ---- decode: out layout is x-major, field-minor (24 contiguous floats) ----
    alignas(16) float o[24];
#pragma unroll
    for (int i = 0; i < 4; ++i) {
        o[6 * i + 0] = (fast_sigmoid(v[0][i]) + (float)(x0 + i)) * sx;   // px
        o[6 * i + 1] = (fast_sigmoid(v[1][i]) + (float)y) * sy;          // py
        o[6 * i + 2] = fast_exp(v[2][i]) * aw;                           // pw
        o[6 * i + 3] = fast_exp(v[3][i]) * ah;                           // ph
        o[6 * i + 4] = fast_sigmoid(v[4][i]) * 360.0f - 180.0f;         // pa
        o[6 * i + 5] = fast_sigmoid(v[5][i]);                            // conf
    }

    // ---- store 96 contiguous bytes as 6x global_store_b128 NT (16B aligned:
    //      base = 96 * quad_index, and 96 % 16 == 0) ----
    float* op = out + (size_t)((b * nA + a) * plane + y * nW + x0) * 6;
    const v4f* ov = (const v4f*)o;
#pragma unroll
    for (int k = 0; k < 6; ++k) {
        __builtin_nontemporal_store(ov[k], (v4f*)op + k);
    }
}

extern "C" void kernel_launch(void* const* d_in, const int* in_sizes, int n_in,
                              void* d_out, int out_size, void* d_ws, size_t ws_size,
                              hipStream_t stream) {
    const float* raw     = (const float*)d_in[0];
    const float* anchors = (const float*)d_in[1];
    const int*   img_h   = (const int*)d_in[2];
    const int*   img_w   = (const int*)d_in[3];
    float*       out     = (float*)d_out;

    constexpr int nB = 64, nA = 3, nH = 128, nW = 128;
    constexpr int quads = nB * nA * nH * nW / 4;   // 786432
    constexpr int block = 256;                     // 8 wave32 waves
    constexpr int grid  = quads / block;           // 3072

    hipLaunchKernelGGL(rapid_decode_kernel, dim3(grid), dim3(block), 0, stream,
                       raw, anchors, img_h, img_w, out);
}